// BodyModel_90812788506907
// MI455X (gfx1250) — compile-verified
//
#include <hip/hip_runtime.h>

// ---------------------------------------------------------------------------
// SMPL-style LBS, fused for MI455X (gfx1250, wave32).
//  B=512 batches, V=6890 verts, J=24 joints, NB=10 betas.
//  Heavy GEMM (pose_feat[512x207] @ posedirs[207x20670]) runs on
//  V_WMMA_F32_16X16X4_F32 (full-precision fp32 WMMA). Everything else is
//  fused so v_shaped / v_posed / per-vertex T never touch HBM.
// ---------------------------------------------------------------------------

#define BATCH 512
#define NV    6890
#define NJ    24
#define NB    10
#define KPOSE 207          // (J-1)*9
#define KPAD  208          // padded to 13 chunks of 16
#define NCOLS (NV * 3)     // 20670
#define CTILE 96           // columns per block (multiple of 3 and 16)
#define VTILE (CTILE / 3)  // 32 vertices per block
#define MTILE 16           // batches per block
#define NWAVE 6            // waves per block (CTILE/16)
#define BLKC  (NWAVE * 32) // 192 threads

typedef __attribute__((ext_vector_type(2))) float v2f;
typedef __attribute__((ext_vector_type(8))) float v8f;

// ---------------------------------------------------------------------------
// Kernel P: Jv = Jreg @ v_template (24x3), Jshape = Jreg @ shapedirs (24x30).
// 792 blocks, each reduces one output over V=6890.
// ---------------------------------------------------------------------------
__global__ __launch_bounds__(256)
void precompute_jreg(const float* __restrict__ Jreg,
                     const float* __restrict__ shapedirs,
                     const float* __restrict__ vtemp,
                     float* __restrict__ Jshape,
                     float* __restrict__ Jv) {
    const int j = blockIdx.x / 33;
    const int o = blockIdx.x % 33;   // 0..29 -> shapedirs col, 30..32 -> v_template col
    __shared__ float red[256];
    float acc = 0.f;
    for (int v = threadIdx.x; v < NV; v += 256) {
        float w = Jreg[j * NV + v];
        float x = (o < 30) ? shapedirs[v * 30 + o] : vtemp[v * 3 + (o - 30)];
        acc += w * x;
    }
    red[threadIdx.x] = acc;
    __syncthreads();
    for (int s = 128; s > 0; s >>= 1) {
        if (threadIdx.x < s) red[threadIdx.x] += red[threadIdx.x + s];
        __syncthreads();
    }
    if (threadIdx.x == 0) {
        if (o < 30) Jshape[j * 30 + o] = red[0];
        else        Jv[j * 3 + (o - 30)] = red[0];
    }
}

// ---------------------------------------------------------------------------
// Kernel B: per-batch small math. One wave (32 threads) per batch.
//  rodrigues -> rot, pose_feat (padded 208), rest joints (via Jv/Jshape),
//  kinematic chain (serial on lane 0), rel_transforms -> ws,
//  posed joints + trans -> d_out.
// ---------------------------------------------------------------------------
__global__ __launch_bounds__(32)
void body_small(const float* __restrict__ root,
                const float* __restrict__ pbody,
                const float* __restrict__ phand,
                const float* __restrict__ betas,
                const float* __restrict__ trans,
                const float* __restrict__ Jshape,
                const float* __restrict__ Jv,
                const int*   __restrict__ parents,
                float* __restrict__ posefeat,   // [B][208]
                float* __restrict__ relT,       // [B][24][12]
                float* __restrict__ out_joints) // [B][24][3]
{
    const int b = blockIdx.x;
    const int t = threadIdx.x;

    __shared__ float rot[NJ][9];
    __shared__ float jnt[NJ][3];
    __shared__ float TsS[NJ][12];    // [R | rel_j] rows (3x4)
    __shared__ float chainS[NJ][12];
    __shared__ int   par[NJ];

    if (t < NJ) {
        par[t] = parents[t];
        // gather axis-angle for joint t
        float r0, r1, r2;
        if (t == 0)       { r0 = root[b*3+0];  r1 = root[b*3+1];  r2 = root[b*3+2]; }
        else if (t <= 21) { int o = b*63 + (t-1)*3; r0 = pbody[o]; r1 = pbody[o+1]; r2 = pbody[o+2]; }
        else              { int o = b*6 + (t-22)*3; r0 = phand[o]; r1 = phand[o+1]; r2 = phand[o+2]; }
        float a2  = r0*r0 + r1*r1 + r2*r2 + 1e-8f;
        float ang = sqrtf(a2);
        float inv = 1.f / ang;
        float x = r0*inv, y = r1*inv, z = r2*inv;
        float c = cosf(ang), s = sinf(ang), oc = 1.f - c;
        rot[t][0] = c + oc*x*x;     rot[t][1] = -s*z + oc*x*y;  rot[t][2] =  s*y + oc*x*z;
        rot[t][3] =  s*z + oc*y*x;  rot[t][4] = c + oc*y*y;     rot[t][5] = -s*x + oc*y*z;
        rot[t][6] = -s*y + oc*z*x;  rot[t][7] =  s*x + oc*z*y;  rot[t][8] = c + oc*z*z;
        // rest-pose joints: Jv + Jshape * betas
        #pragma unroll
        for (int k = 0; k < 3; ++k) {
            float a = Jv[t*3 + k];
            #pragma unroll
            for (int l = 0; l < NB; ++l)
                a += Jshape[(t*3 + k)*NB + l] * betas[b*NB + l];
            jnt[t][k] = a;
        }
    }
    __syncthreads();

    // pose_feat: (rot[1:] - I) flattened, padded with one zero
    for (int i = t; i < KPOSE; i += 32) {
        int j = i / 9 + 1, e = i % 9;
        float d = (e == 0 || e == 4 || e == 8) ? 1.f : 0.f;
        posefeat[b*KPAD + i] = rot[j][e] - d;
    }
    if (t == 0) posefeat[b*KPAD + KPOSE] = 0.f;

    // local transforms Ts = [R | rel_j]
    if (t < NJ) {
        int p = par[t];
        #pragma unroll
        for (int r = 0; r < 3; ++r) {
            TsS[t][r*4+0] = rot[t][r*3+0];
            TsS[t][r*4+1] = rot[t][r*3+1];
            TsS[t][r*4+2] = rot[t][r*3+2];
            TsS[t][r*4+3] = (t == 0) ? jnt[t][r] : (jnt[t][r] - jnt[p][r]);
        }
    }
    __syncthreads();

    // kinematic chain (serial; affine 3x4 compose)
    if (t == 0) {
        #pragma unroll
        for (int e = 0; e < 12; ++e) chainS[0][e] = TsS[0][e];
        for (int j = 1; j < NJ; ++j) {
            const float* A = chainS[par[j]];
            const float* Bm = TsS[j];
            #pragma unroll
            for (int r = 0; r < 3; ++r) {
                #pragma unroll
                for (int c2 = 0; c2 < 3; ++c2) {
                    chainS[j][r*4+c2] = A[r*4+0]*Bm[0*4+c2] + A[r*4+1]*Bm[1*4+c2] + A[r*4+2]*Bm[2*4+c2];
                }
                chainS[j][r*4+3] = A[r*4+0]*Bm[0*4+3] + A[r*4+1]*Bm[1*4+3] + A[r*4+2]*Bm[2*4+3] + A[r*4+3];
            }
        }
    }
    __syncthreads();

    if (t < NJ) {
        // posed joints (+trans) straight to output
        #pragma unroll
        for (int k = 0; k < 3; ++k)
            out_joints[(b*NJ + t)*3 + k] = chainS[t][k*4+3] + trans[b*3 + k];
        // rel transform: subtract chain * [rest_joint, 0] from translation column
        #pragma unroll
        for (int m = 0; m < 3; ++m) {
            float corr = chainS[t][m*4+0]*jnt[t][0] + chainS[t][m*4+1]*jnt[t][1] + chainS[t][m*4+2]*jnt[t][2];
            relT[(b*NJ + t)*12 + m*4 + 0] = chainS[t][m*4+0];
            relT[(b*NJ + t)*12 + m*4 + 1] = chainS[t][m*4+1];
            relT[(b*NJ + t)*12 + m*4 + 2] = chainS[t][m*4+2];
            relT[(b*NJ + t)*12 + m*4 + 3] = chainS[t][m*4+3] - corr;
        }
    }
}

// ---------------------------------------------------------------------------
// Kernel C: fused pose-blend GEMM (fp32 WMMA) + shape blend + skinning.
// Block = 16 batches x 96 columns, 6 waves; each wave owns a 16x16 C tile.
// ---------------------------------------------------------------------------
__global__ __launch_bounds__(BLKC)
void lbs_main(const float* __restrict__ betas,
              const float* __restrict__ trans,
              const float* __restrict__ vtemp,
              const float* __restrict__ shapedirs,
              const float* __restrict__ posedirs,   // [207][20670]
              const float* __restrict__ weights,    // [V][24]
              const float* __restrict__ posefeat,   // [B][208]
              const float* __restrict__ relT,       // [B][24][12]
              float* __restrict__ out_verts)        // [B][V][3]
{
    const int c0  = blockIdx.x * CTILE;
    const int b0  = blockIdx.y * MTILE;
    const int tid = threadIdx.x;
    const int lane = tid & 31;
    const int wave = tid >> 5;

    __shared__ float As[MTILE][KPAD + 1];   // pose_feat tile (pad -> fewer bank conflicts)
    __shared__ float Bs[16][CTILE];         // posedirs K-slab
    __shared__ float vp[MTILE][CTILE];      // v_posed tile
    __shared__ float relTs[MTILE][NJ * 12];
    __shared__ float wts_s[VTILE][NJ];      // weights tile (reused by 16 batches)
    __shared__ float betas_s[MTILE][NB];
    __shared__ float trans_s[MTILE][3];

    // stage per-block constants (once per block; div/mod here is cold)
    for (int i = tid; i < MTILE * KPAD; i += BLKC) {
        int m = i / KPAD, k = i % KPAD;
        As[m][k] = posefeat[(b0 + m) * KPAD + k];
    }
    for (int i = tid; i < MTILE * NB; i += BLKC)
        betas_s[i / NB][i % NB] = betas[(b0 + i / NB) * NB + (i % NB)];
    for (int i = tid; i < MTILE * 3; i += BLKC)
        trans_s[i / 3][i % 3] = trans[(b0 + i / 3) * 3 + (i % 3)];
    for (int i = tid; i < MTILE * NJ * 12; i += BLKC)
        relTs[i / (NJ * 12)][i % (NJ * 12)] = relT[(size_t)(b0 + i / (NJ * 12)) * (NJ * 12) + (i % (NJ * 12))];
    for (int i = tid; i < VTILE * NJ; i += BLKC) {
        int lv = i / NJ, j = i % NJ;
        int v = c0 / 3 + lv;
        wts_s[lv][j] = (v < NV) ? weights[(size_t)v * NJ + j] : 0.f;
    }

    // Fixed per-thread B-staging coordinates: 192 threads cover 2 rows x 96
    // cols per pass, 8 passes per 16-row slab. No div/mod in the hot loop.
    const int  nB    = tid % CTILE;          // column within tile
    const int  kB0   = tid / CTILE;          // 0 or 1
    const int  cgB   = c0 + nB;
    const bool colOK = (cgB < NCOLS);
    const float* pdB = posedirs + cgB;       // column base

    // WMMA fragment coordinates (ISA 7.12.2, 32-bit 16x4 A / 4x16 B / 16x16 C)
    const int mA    = lane & 15;             // A row / B col within tile
    const int khalf = (lane >> 4) << 1;      // lanes 16-31 hold K+2,K+3
    const int wn    = wave * 16 + (lane & 15);

    v8f acc = {0.f, 0.f, 0.f, 0.f, 0.f, 0.f, 0.f, 0.f};

    for (int kc = 0; kc < KPAD; kc += 16) {
        // stage 16 x 96 slab of posedirs (rows contiguous -> coalesced)
        #pragma unroll
        for (int it = 0; it < 8; ++it) {
            const int kk = kB0 + it * 2;
            const int kg = kc + kk;
            float val = 0.f;
            if (colOK && kg < KPOSE) val = pdB[(size_t)kg * NCOLS];
            Bs[kk][nB] = val;
        }
        // hint next slab toward L2 while we compute this one
        if (kc + 16 < KPOSE && colOK)
            __builtin_prefetch(pdB + (size_t)(kc + 16) * NCOLS, 0, 1);
        __syncthreads();

        #pragma unroll
        for (int s = 0; s < 4; ++s) {
            const int ka = kc + s * 4 + khalf;
            const int kb = s * 4 + khalf;
            v2f a, bf;
            a[0]  = As[mA][ka];     a[1]  = As[mA][ka + 1];
            bf[0] = Bs[kb][wn];     bf[1] = Bs[kb + 1][wn];
            acc = __builtin_amdgcn_wmma_f32_16x16x4_f32(
                    false, a, false, bf, (short)0, acc, false, false);
        }
        __syncthreads();
    }

    // epilogue: add v_template + shapedirs*betas, park v_posed in LDS.
    // C layout: lane column = wn; VGPR r -> row r (lanes 0-15) / r+8 (lanes 16-31).
    const int cg = c0 + wn;
    float vtv = 0.f, sdl[NB];
    #pragma unroll
    for (int l = 0; l < NB; ++l) sdl[l] = 0.f;
    if (cg < NCOLS) {
        vtv = vtemp[cg];
        #pragma unroll
        for (int l = 0; l < NB; ++l) sdl[l] = shapedirs[(size_t)cg * NB + l];
    }
    const int mh = (lane >> 4) * 8;
    #pragma unroll
    for (int r = 0; r < 8; ++r) {
        int mm = r + mh;
        float s = acc[r] + vtv;
        #pragma unroll
        for (int l = 0; l < NB; ++l) s += betas_s[mm][l] * sdl[l];
        if (cg < NCOLS) vp[mm][wn] = s;
    }
    __syncthreads();

    // skinning: verts = (sum_j w_j * relT_j) applied to [v_posed, 1], + trans
    const int ncols  = min(CTILE, NCOLS - c0);
    const int nverts = ncols / 3;                 // block column tile is vertex-aligned
    for (int p = tid; p < MTILE * VTILE; p += BLKC) {
        int lv = p & (VTILE - 1), mm = p >> 5;
        if (lv >= nverts) continue;
        int v = c0 / 3 + lv;
        float x = vp[mm][lv * 3 + 0];
        float y = vp[mm][lv * 3 + 1];
        float z = vp[mm][lv * 3 + 2];
        float rx = 0.f, ry = 0.f, rz = 0.f;
        const float* rt = &relTs[mm][0];
        #pragma unroll
        for (int j = 0; j < NJ; ++j) {
            float w = wts_s[lv][j];
            const float* t4 = rt + j * 12;
            rx += w * (t4[0]*x + t4[1]*y + t4[2] *z + t4[3]);
            ry += w * (t4[4]*x + t4[5]*y + t4[6] *z + t4[7]);
            rz += w * (t4[8]*x + t4[9]*y + t4[10]*z + t4[11]);
        }
        size_t o = ((size_t)(b0 + mm) * NV + v) * 3;
        out_verts[o + 0] = rx + trans_s[mm][0];
        out_verts[o + 1] = ry + trans_s[mm][1];
        out_verts[o + 2] = rz + trans_s[mm][2];
    }
}

// ---------------------------------------------------------------------------
// Host launcher
// ---------------------------------------------------------------------------
extern "C" void kernel_launch(void* const* d_in, const int* in_sizes, int n_in,
                              void* d_out, int out_size, void* d_ws, size_t ws_size,
                              hipStream_t stream) {
    const float* root   = (const float*)d_in[0];
    const float* pbody  = (const float*)d_in[1];
    const float* phand  = (const float*)d_in[2];
    const float* betas  = (const float*)d_in[3];
    const float* trans  = (const float*)d_in[4];
    const float* vtemp  = (const float*)d_in[5];
    const float* shaped = (const float*)d_in[6];
    const float* posed  = (const float*)d_in[7];
    const float* Jreg   = (const float*)d_in[8];
    const float* wts    = (const float*)d_in[9];
    const int*   par    = (const int*)  d_in[10];

    // workspace layout (floats): Jshape[720] | Jv[72] | posefeat[512*208] | relT[512*288]
    float* ws       = (float*)d_ws;
    float* Jshape   = ws;
    float* Jv       = ws + 768;
    float* posefeat = ws + 1024;
    float* relT     = ws + 1024 + (size_t)BATCH * KPAD;
    // total ~1.02 MB of d_ws used

    float* out_verts  = (float*)d_out;
    float* out_joints = out_verts + (size_t)BATCH * NV * 3;

    precompute_jreg<<<NJ * 33, 256, 0, stream>>>(Jreg, shaped, vtemp, Jshape, Jv);
    body_small<<<BATCH, 32, 0, stream>>>(root, pbody, phand, betas, trans,
                                         Jshape, Jv, par, posefeat, relT, out_joints);
    dim3 grid((NCOLS + CTILE - 1) / CTILE, BATCH / MTILE);
    lbs_main<<<grid, BLKC, 0, stream>>>(betas, trans, vtemp, shaped, posed, wts,
                                        posefeat, relT, out_verts);
}